// LaneLoss_9216999817598
// MI455X (gfx1250) — compile-verified
//
#include <hip/hip_runtime.h>

// ---------------------------------------------------------------------------
// Lane-detection set loss for MI455X (gfx1250, wave32).
// Pipeline: cost volume (WMMA f32 16x16x4) -> Hungarian (f64, LDS) ->
//           matched-pair losses (WMMA curve routine) -> deterministic finalize.
// ---------------------------------------------------------------------------

constexpr int kB = 16;
constexpr int kQ = 50;
constexpr int kG = 16;
constexpr int kP = 64;
constexpr int kT = 7;

// loss weights
constexpr float EXIST_W  = 1.5f;
constexpr float PTS_W    = 3.0f;
constexpr float TYPE_W   = 0.5f;
constexpr float CURVE_W  = 5.0f;
constexpr float DIR_W    = 1.5f;
constexpr float SMOOTH_W = 0.25f;

typedef __attribute__((ext_vector_type(2))) float v2f;
typedef __attribute__((ext_vector_type(8))) float v8f;

// LDS scratch bundle shared by the curve-distance routine.
struct SegScratch {
  float ABX[kP];   // ab_x per segment
  float ABY[kP];   // ab_y per segment
  float NAB[kP];   // -(a . ab)
  float A2[kP];    // |a|^2  (1e30 poison for column 63)
  float RD[kP];    // 1 / max(|ab|^2, 1e-8)   (0 for column 63)
  float AB2[kP];   // |ab|^2
  float P2[kP];    // |p|^2 per point
};

// ---------------- wave helpers (wave32, fixed-order => deterministic) -------
__device__ __forceinline__ float half16_min(float v) {
#pragma unroll
  for (int m = 1; m < 16; m <<= 1) v = fminf(v, __shfl_xor(v, m, 32));
  return v;
}
__device__ __forceinline__ float wave_sum(float v) {
#pragma unroll
  for (int m = 1; m < 32; m <<= 1) v += __shfl_xor(v, m, 32);
  return v;
}

// ---------------------------------------------------------------------------
// For 64 points P and the 63 segments of curve A, compute per-point min
// point-to-segment distance, return sum_m w[m] * mindist[m] (w==null -> 1),
// replicated across the wave.
//
// Matrix trick: with K=4 rows (p_x, p_y, 1, 0) and columns
//   B_S = (ab_x, ab_y, -a.ab, 0)       ->  S = ap.ab
//   B_R = (-2a_x, -2a_y, |a|^2, 0)     ->  R + |p|^2 = |ap|^2
// both 64x63 matrices come out of V_WMMA_F32_16X16X4_F32 tiles.
// The min runs over squared distances (sqrt is monotonic), so only one
// sqrt per point row.  Column 63 holds a poison segment (d2 ~ 1e30).
// ---------------------------------------------------------------------------
__device__ float min_seg_dist_sum(const float* PX, const float* PY,
                                  const float* AX, const float* AY,
                                  const float* W, SegScratch* sc, int lane) {
  // --- segment + point precompute (lane-parallel, once per direction) ---
  for (int n = lane; n < kP; n += 32) {
    if (n < kP - 1) {
      const float ax = AX[n], ay = AY[n];
      const float abx = AX[n + 1] - ax, aby = AY[n + 1] - ay;
      const float ab2 = abx * abx + aby * aby;
      sc->ABX[n] = abx;
      sc->ABY[n] = aby;
      sc->NAB[n] = -(ax * abx + ay * aby);
      sc->A2[n]  = ax * ax + ay * ay;
      sc->RD[n]  = 1.0f / fmaxf(ab2, 1e-8f);
      sc->AB2[n] = ab2;
    } else {  // poison column: d2 = ~1e30, never wins the min
      sc->ABX[n] = 0.f; sc->ABY[n] = 0.f; sc->NAB[n] = 0.f;
      sc->A2[n] = 1e30f; sc->RD[n] = 0.f; sc->AB2[n] = 0.f;
    }
  }
  for (int m = lane; m < kP; m += 32)
    sc->P2[m] = PX[m] * PX[m] + PY[m] * PY[m];
  __syncthreads();

  const int cl  = lane & 15;
  const int hi8 = (lane >= 16) ? 8 : 0;
  float acc = 0.f;

  for (int Mt = 0; Mt < 4; ++Mt) {
    // A tile: lanes 0-15 -> K=0,1 (coords); lanes 16-31 -> K=2,3 (1, 0)
    v2f Am;
    {
      const int mrow = Mt * 16 + cl;
      if (lane < 16) { Am.x = PX[mrow]; Am.y = PY[mrow]; }
      else           { Am.x = 1.0f;     Am.y = 0.0f;     }
    }

    // Nt-invariant per-row data
    float p2r[8], wr[8], dmin[8];
#pragma unroll
    for (int r = 0; r < 8; ++r) {
      const int m = Mt * 16 + r + hi8;
      p2r[r] = sc->P2[m];
      wr[r]  = W ? W[m] : 1.0f;
      dmin[r] = 3.4e38f;
    }

    for (int Nt = 0; Nt < 4; ++Nt) {
      const int c = Nt * 16 + cl;  // segment (column) for this lane
      v2f Bs, Br;
      if (lane < 16) { Bs.x = sc->ABX[c];       Bs.y = sc->ABY[c];
                       Br.x = -2.f * AX[c];     Br.y = -2.f * AY[c]; }
      else           { Bs.x = sc->NAB[c];       Bs.y = 0.f;
                       Br.x = sc->A2[c];        Br.y = 0.f;          }
      const float rd  = sc->RD[c];
      const float ab2 = sc->AB2[c];

      v8f zero = {};
      v8f S = __builtin_amdgcn_wmma_f32_16x16x4_f32(
          false, Am, false, Bs, (short)0, zero, false, false);
      v8f R = __builtin_amdgcn_wmma_f32_16x16x4_f32(
          false, Am, false, Br, (short)0, zero, false, false);

#pragma unroll
      for (int r = 0; r < 8; ++r) {
        const float Sv = S[r];
        const float Rv = R[r] + p2r[r];                 // |ap|^2
        const float t  = fminf(fmaxf(Sv * rd, 0.f), 1.f);
        const float d2 = Rv - 2.f * t * Sv + t * t * ab2;
        dmin[r] = fminf(dmin[r], d2);
      }
    }

    // reduce each row's squared min across its 16-lane half, sqrt once,
    // weight, accumulate
    float rowsum = 0.f;
#pragma unroll
    for (int r = 0; r < 8; ++r) {
      const float mv = half16_min(dmin[r]);
      rowsum += sqrtf(fmaxf(mv, 1e-24f)) * wr[r];
    }
    acc += rowsum;
  }
  acc += __shfl_xor(acc, 16, 32);  // combine both halves; replicated result
  return acc;
}

// bidirectional curve distance + tangent loss for one (pred, gt, vis) pair
__device__ void pair_curve_dir(const float* PX, const float* PY,
                               const float* GX, const float* GY,
                               const float* VI, SegScratch* sc, int lane,
                               float* curve_out, float* dir_out) {
  float vsum = 0.f;
  for (int m = lane; m < kP; m += 32) vsum += VI[m];
  vsum = wave_sum(vsum);

  const float p2g = min_seg_dist_sum(PX, PY, GX, GY, nullptr, sc, lane) * (1.0f / kP);
  __syncthreads();  // scratch reuse between directions
  const float g2p = min_seg_dist_sum(GX, GY, PX, PY, VI, sc, lane) / fmaxf(vsum, 1.0f);
  *curve_out = 0.5f * (p2g + g2p);

  float num = 0.f, den = 0.f;
  for (int n = lane; n < kP - 1; n += 32) {
    const float ptx = PX[n + 1] - PX[n], pty = PY[n + 1] - PY[n];
    const float gtx = GX[n + 1] - GX[n], gty = GY[n + 1] - GY[n];
    const float pn = fmaxf(sqrtf(ptx * ptx + pty * pty), 1e-6f);
    const float gn = fmaxf(sqrtf(gtx * gtx + gty * gty), 1e-6f);
    const float cosv = (ptx * gtx + pty * gty) / (pn * gn);
    const float sv = VI[n] * VI[n + 1];
    num += (1.f - cosv) * sv;
    den += sv;
  }
  num = wave_sum(num);
  den = wave_sum(den);
  *dir_out = num / fmaxf(den, 1.f);
}

// ---------------------------------------------------------------------------
// Kernel 1: cost[b,q,g] = curve + 0.35*dir - 0.25*sigmoid(exist[b,q])
// ---------------------------------------------------------------------------
__global__ __launch_bounds__(32) void cost_kernel(const float* __restrict__ pred_exist,
                                                  const float* __restrict__ pred_pts,
                                                  const float* __restrict__ gt_pts,
                                                  const float* __restrict__ gt_vis,
                                                  float* __restrict__ cost) {
  __shared__ float PX[kP], PY[kP], GX[kP], GY[kP], VI[kP];
  __shared__ SegScratch sc;
  const int id = blockIdx.x;
  const int b = id / (kQ * kG);
  const int rem = id % (kQ * kG);
  const int q = rem / kG;
  const int g = rem % kG;
  const int lane = threadIdx.x;

  // prefetch the exist logit consumed at the end (global_prefetch_b8)
  __builtin_prefetch(pred_exist + b * kQ + q, 0, 0);

  for (int m = lane; m < kP; m += 32) {
    const float* pp = pred_pts + (((size_t)b * kQ + q) * kP + m) * 2;
    PX[m] = pp[0]; PY[m] = pp[1];
    const float* gp = gt_pts + (((size_t)b * kG + g) * kP + m) * 2;
    GX[m] = gp[0]; GY[m] = gp[1];
    VI[m] = gt_vis[((size_t)b * kG + g) * kP + m];
  }
  __syncthreads();

  float curve, dirl;
  pair_curve_dir(PX, PY, GX, GY, VI, &sc, lane, &curve, &dirl);

  if (lane == 0) {
    const float x = pred_exist[b * kQ + q];
    const float sig = 1.f / (1.f + expf(-x));
    cost[((size_t)b * kQ + q) * kG + g] = curve + 0.35f * dirl - 0.25f * sig;
  }
}

// ---------------------------------------------------------------------------
// Kernel 2: exact Hungarian (Jonker-Volgenant form used by the reference),
// float64 in LDS, one wave per batch (lane 0 serial to match numpy order).
// ---------------------------------------------------------------------------
__global__ __launch_bounds__(32) void hungarian_kernel(const float* __restrict__ cost,
                                                       int* __restrict__ pi,
                                                       int* __restrict__ gi) {
  __shared__ double sq[kQ][kQ];
  __shared__ double u[kQ + 1], v[kQ + 1], minv[kQ + 1];
  __shared__ int p[kQ + 1], way[kQ + 1], r2c[kQ];
  __shared__ unsigned char used[kQ + 1];
  const int b = blockIdx.x;
  const int lane = threadIdx.x;

  // pad = max(cost[b]) + 1
  float mx = -3.4e38f;
  for (int idx = lane; idx < kQ * kG; idx += 32)
    mx = fmaxf(mx, cost[(size_t)b * kQ * kG + idx]);
#pragma unroll
  for (int m = 1; m < 32; m <<= 1) mx = fmaxf(mx, __shfl_xor(mx, m, 32));
  const double pad = (double)mx + 1.0;

  for (int idx = lane; idx < kQ * kQ; idx += 32) {
    const int r = idx / kQ, c = idx % kQ;
    sq[r][c] = (c < kG) ? (double)cost[((size_t)b * kQ + r) * kG + c] : pad;
  }
  __syncthreads();

  if (lane == 0) {
    for (int j = 0; j <= kQ; ++j) { u[j] = 0.0; v[j] = 0.0; p[j] = 0; way[j] = 0; }
    for (int i = 1; i <= kQ; ++i) {
      p[0] = i;
      int j0 = 0;
      for (int j = 0; j <= kQ; ++j) { minv[j] = 1e18; used[j] = 0; }
      for (;;) {
        used[j0] = 1;
        const int i0 = p[j0];
        double delta = 1e18;
        int j1 = -1;
        for (int j = 1; j <= kQ; ++j) {
          if (!used[j]) {
            const double cur = sq[i0 - 1][j - 1] - u[i0] - v[j];
            if (cur < minv[j]) { minv[j] = cur; way[j] = j0; }
            if (minv[j] < delta) { delta = minv[j]; j1 = j; }  // first-min tie-break
          }
        }
        for (int j = 0; j <= kQ; ++j) {
          if (used[j]) { u[p[j]] += delta; v[j] -= delta; }
          else         { minv[j] -= delta; }
        }
        j0 = j1;
        if (p[j0] == 0) break;
      }
      for (;;) {
        const int j1 = way[j0];
        p[j0] = p[j1];
        j0 = j1;
        if (!j0) break;
      }
    }
    for (int j = 1; j <= kQ; ++j) r2c[p[j] - 1] = j - 1;
    int k = 0;
    for (int r = 0; r < kQ; ++r) {
      if (r2c[r] < kG) { pi[b * kG + k] = r; gi[b * kG + k] = r2c[r]; ++k; }
    }
  }
}

// ---------------------------------------------------------------------------
// Kernel 3: per-matched-pair losses, one wave per (b,g) pair.
// pairval = PTS_W*pts + CURVE_W*curve + DIR_W*dir + SMOOTH_W*smooth + TYPE_W*ce
// ---------------------------------------------------------------------------
__global__ __launch_bounds__(32) void pair_kernel(const float* __restrict__ pred_pts,
                                                  const float* __restrict__ pred_type,
                                                  const float* __restrict__ gt_pts,
                                                  const float* __restrict__ gt_vis,
                                                  const int* __restrict__ gt_type,
                                                  const int* __restrict__ pi,
                                                  const int* __restrict__ gi,
                                                  float* __restrict__ pairval) {
  __shared__ float PX[kP], PY[kP], GX[kP], GY[kP], VI[kP];
  __shared__ SegScratch sc;
  const int id = blockIdx.x;
  const int b = id / kG;
  const int g = id % kG;
  const int lane = threadIdx.x;
  const int qi = pi[b * kG + g];
  const int gj = gi[b * kG + g];

  __builtin_prefetch(pred_type + ((size_t)b * kQ + qi) * kT, 0, 0);

  for (int m = lane; m < kP; m += 32) {
    const float* pp = pred_pts + (((size_t)b * kQ + qi) * kP + m) * 2;
    PX[m] = pp[0]; PY[m] = pp[1];
    const float* gp = gt_pts + (((size_t)b * kG + gj) * kP + m) * 2;
    GX[m] = gp[0]; GY[m] = gp[1];
    VI[m] = gt_vis[((size_t)b * kG + gj) * kP + m];
  }
  __syncthreads();

  // vis sum
  float vsum = 0.f;
  for (int m = lane; m < kP; m += 32) vsum += VI[m];
  vsum = wave_sum(vsum);
  const float vden = fmaxf(vsum, 1.f);

  // smooth-L1 points loss
  float pacc = 0.f;
  for (int m = lane; m < kP; m += 32) {
    float s = 0.f;
    {
      const float e = PX[m] - GX[m];
      const float a = fabsf(e);
      s += (a < 1.f) ? 0.5f * e * e : a - 0.5f;
    }
    {
      const float e = PY[m] - GY[m];
      const float a = fabsf(e);
      s += (a < 1.f) ? 0.5f * e * e : a - 0.5f;
    }
    pacc += s * VI[m];
  }
  const float pts = wave_sum(pacc) / vden;

  // curve + tangent (WMMA path)
  float curve, dirl;
  pair_curve_dir(PX, PY, GX, GY, VI, &sc, lane, &curve, &dirl);

  // smoothness (second differences)
  float snum = 0.f, sden = 0.f;
  for (int i = lane; i < kP - 2; i += 32) {
    const float sx = PX[i + 2] - 2.f * PX[i + 1] + PX[i];
    const float sy = PY[i + 2] - 2.f * PY[i + 1] + PY[i];
    const float mag = sqrtf(fmaxf(sx * sx + sy * sy, 1e-24f));
    const float cv = VI[i] * VI[i + 1] * VI[i + 2];
    snum += mag * cv;
    sden += cv;
  }
  snum = wave_sum(snum);
  sden = wave_sum(sden);
  const float smooth = snum / fmaxf(sden, 1.f);

  if (lane == 0) {
    // cross-entropy on the matched type logits
    const float* lg = pred_type + ((size_t)b * kQ + qi) * kT;
    const int lbl = gt_type[b * kG + g];
    float mxl = lg[0];
    for (int t = 1; t < kT; ++t) mxl = fmaxf(mxl, lg[t]);
    float se = 0.f;
    for (int t = 0; t < kT; ++t) se += expf(lg[t] - mxl);
    const float ce = -(lg[lbl] - mxl - logf(se));

    pairval[id] = PTS_W * pts + CURVE_W * curve + DIR_W * dirl +
                  SMOOTH_W * smooth + TYPE_W * ce;
  }
}

// ---------------------------------------------------------------------------
// Kernel 4: BCE over existence logits + final weighted combine (one wave,
// fixed reduction order => deterministic).
// ---------------------------------------------------------------------------
__global__ __launch_bounds__(32) void finalize_kernel(const float* __restrict__ pred_exist,
                                                      const int* __restrict__ pi,
                                                      const float* __restrict__ pairval,
                                                      float* __restrict__ out) {
  const int lane = threadIdx.x;
  float bces = 0.f;
  for (int idx = lane; idx < kB * kQ; idx += 32) {
    const int b = idx / kQ;
    const int q = idx % kQ;
    float tgt = 0.f;
    for (int g = 0; g < kG; ++g)
      if (pi[b * kG + g] == q) tgt = 1.f;
    const float x = pred_exist[idx];
    bces += fmaxf(x, 0.f) - x * tgt + log1pf(expf(-fabsf(x)));
  }
  bces = wave_sum(bces);

  float ps = 0.f;
  for (int idx = lane; idx < kB * kG; idx += 32) ps += pairval[idx];
  ps = wave_sum(ps);

  if (lane == 0)
    out[0] = EXIST_W * (bces / (float)(kB * kQ)) + ps / (float)(kB * kG);
}

// ---------------------------------------------------------------------------
extern "C" void kernel_launch(void* const* d_in, const int* in_sizes, int n_in,
                              void* d_out, int out_size, void* d_ws, size_t ws_size,
                              hipStream_t stream) {
  (void)in_sizes; (void)n_in; (void)out_size; (void)ws_size;

  const float* pred_exist = (const float*)d_in[0];  // (B,Q,1)
  const float* pred_pts   = (const float*)d_in[1];  // (B,Q,P,2)
  const float* pred_type  = (const float*)d_in[2];  // (B,Q,7)
  // d_in[3] gt_existence: unused by the reference
  const float* gt_points  = (const float*)d_in[4];  // (B,G,P,2)
  const float* gt_vis     = (const float*)d_in[5];  // (B,G,P)
  const int*   gt_type    = (const int*)d_in[6];    // (B,G) int32
  // d_in[7] has_lanes: unused by the reference
  float* out = (float*)d_out;

  // workspace layout (all rewritten each call)
  float* cost = (float*)d_ws;                                   // B*Q*G
  int*   pi   = (int*)((char*)d_ws + sizeof(float) * kB * kQ * kG);
  int*   gi   = pi + kB * kG;
  float* pairval = (float*)(gi + kB * kG);                      // B*G

  cost_kernel<<<dim3(kB * kQ * kG), dim3(32), 0, stream>>>(
      pred_exist, pred_pts, gt_points, gt_vis, cost);
  hungarian_kernel<<<dim3(kB), dim3(32), 0, stream>>>(cost, pi, gi);
  pair_kernel<<<dim3(kB * kG), dim3(32), 0, stream>>>(
      pred_pts, pred_type, gt_points, gt_vis, gt_type, pi, gi, pairval);
  finalize_kernel<<<dim3(1), dim3(32), 0, stream>>>(pred_exist, pi, pairval, out);
}